// NSED_71657234366496
// MI455X (gfx1250) — compile-verified
//
#include <hip/hip_runtime.h>
#include <stdint.h>

// ---------------------------------------------------------------------------
// NSED gcn_propagate for MI455X (gfx1250).
// Per behavior: build packed CSR once (count -> scan -> scatter), then 3 fused
// gather-SpMM layers (wave-per-row, zero float atomics, accumulator fused),
// then fused L2-normalize + residual. Edge records staged via CDNA5 async
// global->LDS (ASYNCcnt path). wave32 throughout.
// ---------------------------------------------------------------------------

#define N_USERS 100000
#define N_ITEMS 100000
#define EMB     64
#define NNODES  (N_USERS + N_ITEMS + 2)   // 200002
#define NEDGES  1200000
#define NB      3
#define RPB     8                          // rows per block in spmm
#define CAP     1024                       // LDS edge capacity per spmm block
#define NBLK    ((NNODES + 255) / 256)     // 782 scan blocks

constexpr size_t ND  = (size_t)NNODES * EMB;   // 12,800,128 floats (51.2 MB)
constexpr size_t ND4 = ND / 4;                 // 3,200,032 float4s

// ---- CDNA5 async global->LDS helpers (ASYNCcnt path) ----------------------
typedef __attribute__((address_space(1))) int  g_int;
typedef __attribute__((address_space(3))) int  l_int;

template <int OFF>
__device__ __forceinline__ void async_g2l_b32(const void* g, void* l) {
#if defined(__gfx1250__) && __has_builtin(__builtin_amdgcn_global_load_async_to_lds_b32)
  // INST_OFFSET is applied to both the global and the LDS address (ISA 10.x).
  __builtin_amdgcn_global_load_async_to_lds_b32(
      (g_int*)(uintptr_t)g,
      (l_int*)(uint32_t)(uintptr_t)l,
      OFF, /*cpol*/0);
#else
  unsigned lds_off = (unsigned)(uintptr_t)l + (unsigned)OFF;
  const char* gp = (const char*)g + OFF;
  asm volatile("global_load_async_to_lds_b32 %0, %1, off"
               :: "v"(lds_off), "v"(gp) : "memory");
#endif
}

__device__ __forceinline__ void wait_async0() {
#if defined(__gfx1250__) && __has_builtin(__builtin_amdgcn_s_wait_asynccnt)
  __builtin_amdgcn_s_wait_asynccnt(0);
#else
  asm volatile("s_wait_asynccnt 0" ::: "memory");
#endif
}

// ---- CSR build ------------------------------------------------------------
__global__ void __launch_bounds__(256) zero_int_k(int* __restrict__ p, int n) {
  int i = blockIdx.x * 256 + threadIdx.x;
  if (i < n) p[i] = 0;
}

__global__ void __launch_bounds__(256) count_k(const int* __restrict__ rows,
                                               int* __restrict__ cnt) {
  int e = blockIdx.x * 256 + threadIdx.x;
  if (e < NEDGES) atomicAdd(&cnt[rows[e]], 1);
}

__global__ void __launch_bounds__(256) block_sums_k(const int* __restrict__ cnt,
                                                    int* __restrict__ bsum) {
  __shared__ int sm[256];
  const int t = threadIdx.x;
  const int i = blockIdx.x * 256 + t;
  sm[t] = (i < NNODES) ? cnt[i] : 0;
  __syncthreads();
  for (int off = 128; off > 0; off >>= 1) {
    if (t < off) sm[t] += sm[t + off];
    __syncthreads();
  }
  if (t == 0) bsum[blockIdx.x] = sm[0];
}

__global__ void __launch_bounds__(1024) scan_sums_k(const int* __restrict__ bsum,
                                                    int* __restrict__ boff, int nb) {
  __shared__ int sm[1024];
  const int t = threadIdx.x;
  const int v = (t < nb) ? bsum[t] : 0;
  sm[t] = v;
  __syncthreads();
  for (int off = 1; off < 1024; off <<= 1) {
    int x = (t >= off) ? sm[t - off] : 0;
    __syncthreads();
    sm[t] += x;
    __syncthreads();
  }
  if (t < nb) boff[t] = sm[t] - v;   // exclusive prefix of block sums
}

__global__ void __launch_bounds__(256) scan_final_k(const int* __restrict__ cnt,
                                                    const int* __restrict__ boff,
                                                    int* __restrict__ row_ptr,
                                                    int* __restrict__ fill) {
  __shared__ int sm[256];
  const int t = threadIdx.x;
  const int i = blockIdx.x * 256 + t;
  const int v = (i < NNODES) ? cnt[i] : 0;
  sm[t] = v;
  __syncthreads();
  for (int off = 1; off < 256; off <<= 1) {
    int x = (t >= off) ? sm[t - off] : 0;
    __syncthreads();
    sm[t] += x;
    __syncthreads();
  }
  if (i < NNODES) {
    const int excl = sm[t] - v + boff[blockIdx.x];
    row_ptr[i] = excl;
    fill[i]    = excl;   // scatter cursor
  }
  if (i == 0) row_ptr[NNODES] = NEDGES;
}

// packed CSR record: .x = col, .y = bit-pattern of weight
__global__ void __launch_bounds__(256) scatter_k(const int* __restrict__ rows,
                                                 const int* __restrict__ cols,
                                                 const float* __restrict__ wts,
                                                 int* __restrict__ fill,
                                                 int2* __restrict__ csr_pk) {
  int e = blockIdx.x * 256 + threadIdx.x;
  if (e >= NEDGES) return;
  const int r = rows[e];
  const int pos = atomicAdd(&fill[r], 1);
  int2 rec;
  rec.x = cols[e];
  rec.y = __float_as_int(wts[e]);
  csr_pk[pos] = rec;              // single 8 B random store
}

// ---- elementwise init -----------------------------------------------------
__global__ void __launch_bounds__(256) init_concat_k(const float* __restrict__ ue,
                                                     const float* __restrict__ ie,
                                                     float* __restrict__ X,
                                                     float* __restrict__ A) {
  size_t i = (size_t)blockIdx.x * 256 + threadIdx.x;
  if (i >= ND4) return;
  size_t node = i / (EMB / 4);
  size_t sub  = i % (EMB / 4);
  float4 v = (node <= (size_t)N_USERS)
      ? ((const float4*)ue)[node * (EMB / 4) + sub]
      : ((const float4*)ie)[(node - (N_USERS + 1)) * (EMB / 4) + sub];
  ((float4*)X)[i] = v;
  ((float4*)A)[i] = v;
}

__global__ void __launch_bounds__(256) init_copy2_k(const float* __restrict__ src,
                                                    float* __restrict__ X,
                                                    float* __restrict__ A) {
  size_t i = (size_t)blockIdx.x * 256 + threadIdx.x;
  if (i >= ND4) return;
  float4 v = ((const float4*)src)[i];
  ((float4*)X)[i] = v;
  ((float4*)A)[i] = v;
}

// ---- fused CSR SpMM: Y[r] = sum_j w_j * X[col_j];  A[r] += Y[r] -----------
// One wave per row, float2 per lane (256 B coalesced gather per edge).
// Block's contiguous packed-edge span staged to LDS via async global->LDS.
__global__ void __launch_bounds__(256)
spmm_csr_k(const int* __restrict__ row_ptr, const int2* __restrict__ csr_pk,
           const float* __restrict__ X, float* __restrict__ Y,
           float* __restrict__ A) {
  __shared__ int2 s_pk[CAP];
  __shared__ int  s_bounds[RPB + 1];

  const int t    = threadIdx.x;
  const int row0 = blockIdx.x * RPB;

  if (t <= RPB) {
    int rr = row0 + t;
    s_bounds[t] = row_ptr[rr > NNODES ? NNODES : rr];
  }
  __syncthreads();

  const int blkS = s_bounds[0];
  const int nE   = s_bounds[RPB] - blkS;
  const bool useLds = (nE <= CAP);
  if (useLds) {
    for (int j = t; j < nE; j += 256) {
      async_g2l_b32<0>(csr_pk + blkS + j, &s_pk[j]);   // col
      async_g2l_b32<4>(csr_pk + blkS + j, &s_pk[j]);   // weight bits
    }
    wait_async0();          // this wave's async writes landed in LDS
  }
  __syncthreads();          // all waves' staging complete

  const int wave = t >> 5;
  const int lane = t & 31;
  const int row  = row0 + wave;
  if (row >= NNODES) return;

  const int rs = s_bounds[wave];
  const int re = s_bounds[wave + 1];

  float ax = 0.f, ay = 0.f;
  for (int j = rs; j < re; ++j) {
    const int2 rec = useLds ? s_pk[j - blkS] : csr_pk[j];  // ds_load_b64 bcast
    const int   c = rec.x;
    const float w = __int_as_float(rec.y);
    const float2 v = ((const float2*)(X + (size_t)c * EMB))[lane];
    ax += v.x * w;
    ay += v.y * w;
  }

  float2 yv; yv.x = ax; yv.y = ay;
  ((float2*)(Y + (size_t)row * EMB))[lane] = yv;

  float2* ap = (float2*)(A + (size_t)row * EMB) + lane;
  float2 av = *ap;
  av.x += ax; av.y += ay;
  *ap = av;
}

// ---- fused: out = normalize(A/4) + total_prev (in place on A slot) --------
__global__ void __launch_bounds__(256)
norm_residual_k(const float* __restrict__ A, const float* __restrict__ tprev,
                const float* __restrict__ ue, const float* __restrict__ ie,
                float* __restrict__ out, int use_concat) {
  const int node = blockIdx.x * 8 + (threadIdx.x >> 5);
  const int lane = threadIdx.x & 31;
  if (node >= NNODES) return;

  const float2 a = ((const float2*)(A + (size_t)node * EMB))[lane];
  const float vx = a.x * 0.25f;   // acc / (layers+1)
  const float vy = a.y * 0.25f;
  float p = vx * vx + vy * vy;
  for (int off = 16; off > 0; off >>= 1) p += __shfl_xor(p, off, 32);
  const float scale = 1.0f / fmaxf(sqrtf(p), 1e-12f);

  float2 tp;
  if (use_concat) {
    tp = (node <= N_USERS)
        ? ((const float2*)(ue + (size_t)node * EMB))[lane]
        : ((const float2*)(ie + (size_t)(node - (N_USERS + 1)) * EMB))[lane];
  } else {
    tp = ((const float2*)(tprev + (size_t)node * EMB))[lane];
  }

  float2 o;
  o.x = vx * scale + tp.x;
  o.y = vy * scale + tp.y;
  ((float2*)(out + (size_t)node * EMB))[lane] = o;
}

// ---------------------------------------------------------------------------
extern "C" void kernel_launch(void* const* d_in, const int* in_sizes, int n_in,
                              void* d_out, int out_size, void* d_ws, size_t ws_size,
                              hipStream_t stream) {
  (void)in_sizes; (void)n_in; (void)out_size; (void)ws_size;

  const float* ue   = (const float*)d_in[0];   // [100001, 64]
  const float* ie   = (const float*)d_in[1];   // [100001, 64]
  const int*   rows = (const int*)  d_in[2];   // [3, E]
  const int*   cols = (const int*)  d_in[3];   // [3, E]
  const float* wts  = (const float*)d_in[4];   // [3, E]
  float* out = (float*)d_out;                  // [3, N, 64]

  // workspace layout (~114 MB), csr_pk placed after X, Y and before the int arrays for 8 B alignment
  float* X       = (float*)d_ws;                 // ND floats
  float* Y       = X + ND;                       // ND floats
  int2*  csr_pk  = (int2*)(Y + ND);              // NEDGES int2 (8 B aligned)
  int*   row_ptr = (int*)(csr_pk + NEDGES);      // NNODES+1
  int*   fill    = row_ptr + NNODES + 1;         // NNODES
  int*   cnt     = fill + NNODES;                // NNODES
  int*   bsum    = cnt + NNODES;                 // NBLK (padded 1024)
  int*   boff    = bsum + 1024;                  // NBLK (padded 1024)

  const dim3 blk(256);
  const int gElem = (int)((ND4 + 255) / 256);   // 12501
  const int gEdge = (NEDGES + 255) / 256;       // 4688
  const int gNode = NBLK;                       // 782
  const int gRow  = (NNODES + RPB - 1) / RPB;   // 25001
  const int gNorm = (NNODES + 7) / 8;           // 25001

  for (int b = 0; b < NB; ++b) {
    const int*   rows_b = rows + (size_t)b * NEDGES;
    const int*   cols_b = cols + (size_t)b * NEDGES;
    const float* wts_b  = wts  + (size_t)b * NEDGES;
    float* Acc = out + (size_t)b * ND;   // accumulator lives in the output slot

    // ---- build packed CSR for this behavior (amortized over 3 layers) ----
    zero_int_k  <<<gNode, blk, 0, stream>>>(cnt, NNODES);
    count_k     <<<gEdge, blk, 0, stream>>>(rows_b, cnt);
    block_sums_k<<<gNode, blk, 0, stream>>>(cnt, bsum);
    scan_sums_k <<<1, 1024, 0, stream>>>(bsum, boff, NBLK);
    scan_final_k<<<gNode, blk, 0, stream>>>(cnt, boff, row_ptr, fill);
    scatter_k   <<<gEdge, blk, 0, stream>>>(rows_b, cols_b, wts_b, fill, csr_pk);

    // ---- init total into X and the accumulator slot ----
    if (b == 0)
      init_concat_k<<<gElem, blk, 0, stream>>>(ue, ie, X, Acc);
    else
      init_copy2_k<<<gElem, blk, 0, stream>>>(out + (size_t)(b - 1) * ND, X, Acc);

    // ---- 3 LightGCN layers: Y = S@X, Acc += Y, swap ----
    for (int l = 0; l < 3; ++l) {
      spmm_csr_k<<<gRow, blk, 0, stream>>>(row_ptr, csr_pk, X, Y, Acc);
      float* tmp = X; X = Y; Y = tmp;
    }

    // ---- total = normalize(Acc/4) + total_prev ----
    norm_residual_k<<<gNorm, blk, 0, stream>>>(
        Acc, (b == 0) ? (const float*)nullptr : out + (size_t)(b - 1) * ND,
        ue, ie, out + (size_t)b * ND, (b == 0) ? 1 : 0);
  }
}